// OverISS_T_2_13675175871174
// MI455X (gfx1250) — compile-verified
//
#include <hip/hip_runtime.h>
#include <hip/hip_bf16.h>

#define NT 256
constexpr int NB = 2, NCH = 6, NSRC = 4, NBAK = 2, NFRQ = 320, NFR = 500, NTAPS = 4;
constexpr int NFP = 512; // padded frame count (zero-filled tail) -> branch-free WMMA frags
constexpr int NBF = NB * NFRQ;
constexpr float EPS = 1e-3f, TINY = 1e-20f, BEPS = 1e-5f;
constexpr float NORM = 1.0f / (float)NFR;

// workspace layout (floats)
constexpr size_t OFF_Z   = 0;
constexpr size_t OFF_W   = OFF_Z + (size_t)NBF * NBAK * NFR * 2;        // Z: re/im
constexpr size_t OFF_H   = OFF_W + (size_t)NBF * NSRC * NCH * 2;
constexpr size_t OFF_A   = OFF_H + (size_t)NBF * NSRC * NCH * NTAPS * 2;
constexpr size_t OFF_J   = OFF_A + (size_t)NBF * NSRC * NSRC * 2;
constexpr size_t OFF_WGT = OFF_J + (size_t)NBF * NBAK * NSRC * 2;       // w: real, B*4*500

typedef float v2f __attribute__((ext_vector_type(2)));
typedef float v8f __attribute__((ext_vector_type(8)));

#define WMMA_F32X4(A, B, C) \
  __builtin_amdgcn_wmma_f32_16x16x4_f32(false, (A), false, (B), (short)0, (C), false, false)

struct cpx { float r, i; };
__device__ __forceinline__ cpx cmul(cpx a, cpx b) { return {a.r*b.r - a.i*b.i, a.r*b.i + a.i*b.r}; }
__device__ __forceinline__ cpx csub(cpx a, cpx b) { return {a.r - b.r, a.i - b.i}; }
__device__ __forceinline__ cpx crec(cpx b) { float m = b.r*b.r + b.i*b.i; return {b.r/m, -b.i/m}; }

// wave32 shuffle reduce + one LDS float atomic per wave (ds_add_f32)
__device__ __forceinline__ void blockReduceAdd(float* dst, float v) {
  #pragma unroll
  for (int o = 16; o > 0; o >>= 1) v += __shfl_down(v, o, 32);
  if ((threadIdx.x & 31) == 0) atomicAdd(dst, v);
}

// ---------------------------------------------------------------------------
// Kernel 1: per-(b,f) init. C_XX via f32 WMMA, 4x4 complex solve -> J,
// Y = X[:4] -> d_out, Z = J*X[:4] - X[4:] via WMMA, state -> ws.
// ---------------------------------------------------------------------------
__global__ __launch_bounds__(NT) void init_kernel(const float* __restrict__ Xre,
                                                  const float* __restrict__ Xim,
                                                  float* __restrict__ out,
                                                  float* __restrict__ ws) {
  __shared__ float sXr[NCH][NFP], sXi[NCH][NFP];
  __shared__ float zrow[NFP]; // all-zero row for out-of-range matrix rows
  __shared__ float sZr[NBAK][NFR], sZi[NBAK][NFR];
  __shared__ float sCr[NSRC][NCH], sCi[NSRC][NCH];
  __shared__ float sJr[NBAK][NSRC], sJi[NBAK][NSRC];

  const int tid = threadIdx.x;
  const int bf = blockIdx.x;
  const int b = bf / NFRQ, f = bf % NFRQ;

  if (tid == 0) { // emits global_prefetch_b8
    __builtin_prefetch(&Xre[(((size_t)b * NCH) * NFRQ + f) * (size_t)NFR], 0, 1);
    __builtin_prefetch(&Xim[(((size_t)b * NCH) * NFRQ + f) * (size_t)NFR], 0, 1);
  }
  for (int idx = tid; idx < NCH * NFP; idx += NT) {
    int c = idx / NFP, n = idx % NFP;
    if (n < NFR) {
      size_t g = (((size_t)b * NCH + c) * NFRQ + f) * (size_t)NFR + n;
      sXr[c][n] = Xre[g];
      sXi[c][n] = Xim[g];
    } else {
      sXr[c][n] = 0.f;
      sXi[c][n] = 0.f;
    }
  }
  for (int n = tid; n < NFP; n += NT) zrow[n] = 0.f;
  if (tid < NSRC * NCH) { sCr[tid / NCH][tid % NCH] = 0.f; sCi[tid / NCH][tid % NCH] = 0.f; }
  __syncthreads();

  // ---- C_XX[s][d] = (1/N) sum_n X[s][n] conj(X[d][n]) via 4 real WMMA chains.
  // Frag layout (wave32, V_WMMA_F32_16X16X4_F32):
  //  A 16x4: lane L row m=L&15; VGPR{0,1} = K{0,1} (lanes<16) / K{2,3} (lanes>=16)
  //  B 4x16: lane L col n=L&15; VGPR{0,1} = K{0,1} (lanes<16) / K{2,3} (lanes>=16)
  //  D 16x16: lane L col n=L&15; VGPR v row m = v + (L<16 ? 0 : 8)
  // Branch-free: row m>=6 reads the zeroed LDS row via loop-invariant pointer
  // select; K range never exceeds 499 (and tail is zero-padded to 512 anyway).
  {
    const int lane = tid & 31, wave = tid >> 5;
    const int m = lane & 15;
    const int kh = (lane < 16) ? 0 : 2;
    const float* rowR = (m < NCH) ? &sXr[m][0] : &zrow[0];
    const float* rowI = (m < NCH) ? &sXi[m][0] : &zrow[0];
    v8f aRR = {0.f,0.f,0.f,0.f,0.f,0.f,0.f,0.f};
    v8f aII = aRR, aIR = aRR, aRI = aRR;
    for (int n0 = wave * 4; n0 < NFR; n0 += (NT / 32) * 4) {
      const int k0 = n0 + kh;
      v2f Ar, Ai;
      Ar.x = rowR[k0];
      Ar.y = rowR[k0 + 1];
      Ai.x = rowI[k0];
      Ai.y = rowI[k0 + 1];
      // B col index == lane&15 == m and B[k][n] = X[n][k]: B frag == A frag per lane
      aRR = WMMA_F32X4(Ar, Ar, aRR);
      aII = WMMA_F32X4(Ai, Ai, aII);
      aIR = WMMA_F32X4(Ai, Ar, aIR);
      aRI = WMMA_F32X4(Ar, Ai, aRI);
    }
    #pragma unroll
    for (int vv = 0; vv < 8; ++vv) {
      const int mm = vv + ((lane < 16) ? 0 : 8);
      const int nn = lane & 15;
      if (mm < NSRC && nn < NCH) {
        atomicAdd(&sCr[mm][nn], (aRR[vv] + aII[vv]) * NORM); // Re = rr + ii
        atomicAdd(&sCi[mm][nn], (aIR[vv] - aRI[vv]) * NORM); // Im = ir - ri
      }
    }
  }
  __syncthreads();

  // ---- background_update specialized to W=eye, H=0: solve (C[:4,:4]+load) J_H = C[:4,4:6]
  if (tid == 0) {
    cpx M[NSRC][NSRC + NBAK];
    for (int s = 0; s < NSRC; ++s)
      for (int d = 0; d < NSRC + NBAK; ++d) M[s][d] = {sCr[s][d], sCi[s][d]};
    float load = 0.f;
    for (int s = 0; s < NSRC; ++s) load += sqrtf(M[s][s].r * M[s][s].r + M[s][s].i * M[s][s].i);
    load = fmaxf(load * BEPS, BEPS);
    for (int s = 0; s < NSRC; ++s) M[s][s].r += load;
    for (int k = 0; k < NSRC; ++k) {
      int p = k; float best = M[k][k].r * M[k][k].r + M[k][k].i * M[k][k].i;
      for (int r = k + 1; r < NSRC; ++r) {
        float mg = M[r][k].r * M[r][k].r + M[r][k].i * M[r][k].i;
        if (mg > best) { best = mg; p = r; }
      }
      if (p != k) for (int j = 0; j < NSRC + NBAK; ++j) { cpx t = M[k][j]; M[k][j] = M[p][j]; M[p][j] = t; }
      cpx inv = crec(M[k][k]);
      for (int j = 0; j < NSRC + NBAK; ++j) M[k][j] = cmul(M[k][j], inv);
      for (int r = 0; r < NSRC; ++r) if (r != k) {
        cpx fc = M[r][k];
        for (int j = 0; j < NSRC + NBAK; ++j) M[r][j] = csub(M[r][j], cmul(fc, M[k][j]));
      }
    }
    for (int e = 0; e < NBAK; ++e)
      for (int s = 0; s < NSRC; ++s) { sJr[e][s] = M[s][NSRC + e].r; sJi[e][s] = -M[s][NSRC + e].i; }
  }
  __syncthreads();

  // ---- Y init = X[:4] -> d_out (interleaved complex64 layout)
  for (int idx = tid; idx < NSRC * NFR; idx += NT) {
    int c = idx / NFR, n = idx % NFR;
    size_t g = (((size_t)b * NSRC + c) * NFRQ + f) * (size_t)NFR + n;
    out[2 * g] = sXr[c][n];
    out[2 * g + 1] = sXi[c][n];
  }

  // ---- Z[e][n] = sum_d J[e][d] X[d][n] - X[4+e][n] via WMMA (M=2,K=4,N=16 tiles)
  // B-frag loads are branch-free: n1 <= 511 and X tail is zero-padded.
  {
    const int lane = tid & 31, wave = tid >> 5;
    const int m = lane & 15;
    const int kh = (lane < 16) ? 0 : 2;
    v2f AJr, AJi;
    AJr.x = (m < NBAK) ? sJr[m][kh] : 0.f;
    AJr.y = (m < NBAK) ? sJr[m][kh + 1] : 0.f;
    AJi.x = (m < NBAK) ? sJi[m][kh] : 0.f;
    AJi.y = (m < NBAK) ? sJi[m][kh + 1] : 0.f;
    const v8f Z0 = {0.f,0.f,0.f,0.f,0.f,0.f,0.f,0.f};
    for (int n0 = wave * 16; n0 < NFR; n0 += (NT / 32) * 16) {
      const int n1 = n0 + (lane & 15);
      v2f Bxr, Bxi;
      Bxr.x = sXr[kh][n1];
      Bxr.y = sXr[kh + 1][n1];
      Bxi.x = sXi[kh][n1];
      Bxi.y = sXi[kh + 1][n1];
      v8f aRR = Z0, aII = Z0, aRI = Z0, aIR = Z0;
      aRR = WMMA_F32X4(AJr, Bxr, aRR);
      aII = WMMA_F32X4(AJi, Bxi, aII);
      aRI = WMMA_F32X4(AJr, Bxi, aRI);
      aIR = WMMA_F32X4(AJi, Bxr, aIR);
      if (lane < 16 && n1 < NFR) {
        #pragma unroll
        for (int e = 0; e < NBAK; ++e) { // D row m=e is acc VGPR e, lanes 0..15
          sZr[e][n1] = aRR[e] - aII[e] - sXr[NSRC + e][n1];
          sZi[e][n1] = aRI[e] + aIR[e] - sXi[NSRC + e][n1];
        }
      }
    }
  }
  __syncthreads();

  // ---- persist state
  for (int idx = tid; idx < NBAK * NFR; idx += NT) {
    size_t o = OFF_Z + ((size_t)bf * NBAK * NFR + idx) * 2;
    ws[o] = sZr[idx / NFR][idx % NFR];
    ws[o + 1] = sZi[idx / NFR][idx % NFR];
  }
  if (tid < NSRC * NCH) { // W = eye(4,6)
    size_t o = OFF_W + ((size_t)bf * NSRC * NCH + tid) * 2;
    ws[o] = ((tid / NCH) == (tid % NCH)) ? 1.f : 0.f; ws[o + 1] = 0.f;
  }
  for (int idx = tid; idx < NSRC * NCH * NTAPS; idx += NT) { // H = 0
    size_t o = OFF_H + ((size_t)bf * NSRC * NCH * NTAPS + idx) * 2;
    ws[o] = 0.f; ws[o + 1] = 0.f;
  }
  if (tid < NSRC * NSRC) { // A = eye(4)
    size_t o = OFF_A + ((size_t)bf * NSRC * NSRC + tid) * 2;
    ws[o] = ((tid / NSRC) == (tid % NSRC)) ? 1.f : 0.f; ws[o + 1] = 0.f;
  }
  if (tid < NBAK * NSRC) {
    size_t o = OFF_J + ((size_t)bf * NBAK * NSRC + tid) * 2;
    ws[o] = sJr[tid / NSRC][tid % NSRC]; ws[o + 1] = sJi[tid / NSRC][tid % NSRC];
  }
}

// ---------------------------------------------------------------------------
// Kernel 2: w[b,s,n] = 1 / max(mean_f |Y[b,s,f,n]|^2, EPS)  (cross-freq reduce)
// ---------------------------------------------------------------------------
__global__ __launch_bounds__(NT) void weight_kernel(const float* __restrict__ out,
                                                    float* __restrict__ ws) {
  const int bs = blockIdx.x; // b*NSRC + s
  for (int n = threadIdx.x; n < NFR; n += NT) {
    float s = 0.f;
    const float* base = out + ((size_t)bs * NFRQ) * (size_t)NFR * 2 + (size_t)n * 2;
    for (int f = 0; f < NFRQ; ++f) {
      float re = base[(size_t)f * NFR * 2];
      float im = base[(size_t)f * NFR * 2 + 1];
      s += re * re + im * im;
    }
    ws[OFF_WGT + (size_t)bs * NFR + n] = 1.f / fmaxf(s * (1.f / (float)NFRQ), EPS);
  }
}

// ---------------------------------------------------------------------------
// Kernel 3: full LDS-resident ISS sweep for one (b,f).
// ---------------------------------------------------------------------------
__global__ __launch_bounds__(NT) void iss_kernel(const float* __restrict__ Xre,
                                                 const float* __restrict__ Xim,
                                                 float* __restrict__ out,
                                                 float* __restrict__ ws) {
  __shared__ float sXr[NCH][NFR], sXi[NCH][NFR];
  __shared__ float sYr[NSRC][NFR], sYi[NSRC][NFR];
  __shared__ float sZr[NBAK][NFR], sZi[NBAK][NFR];
  __shared__ float sw[NSRC][NFR];
  __shared__ float sWr[NSRC][NCH], sWi[NSRC][NCH];
  __shared__ float sHr[NSRC][NCH][NTAPS], sHi[NSRC][NCH][NTAPS];
  __shared__ float sAr[NSRC][NSRC], sAi[NSRC][NSRC];
  __shared__ float sJr[NBAK][NSRC], sJi[NBAK][NSRC];
  __shared__ float acc[20];
  __shared__ float svr[NCH], svi[NCH];
  __shared__ float sgr[NCH], sgi[NCH];
  __shared__ float spr[NCH], spi[NCH], sdd[NCH];
  __shared__ float sg2r[NSRC], sg2i[NSRC], sv2r[NSRC], sv2i[NSRC];
  __shared__ float sgAr[NSRC], sgAi[NSRC], sAvr[NSRC], sAvi[NSRC];
  __shared__ float sWsr[NCH], sWsi[NCH];
  __shared__ float sHsr[NCH * NTAPS], sHsi[NCH * NTAPS];
  __shared__ float sur[NSRC], sui[NSRC];
  __shared__ float smul[2];
  __shared__ float sBr[NSRC][NSRC], sBi[NSRC][NSRC];

  const int tid = threadIdx.x;
  const int bf = blockIdx.x;
  const int b = bf / NFRQ, f = bf % NFRQ;

  // ---- load state ----
  for (int idx = tid; idx < NCH * NFR; idx += NT) {
    int c = idx / NFR, n = idx % NFR;
    size_t g = (((size_t)b * NCH + c) * NFRQ + f) * (size_t)NFR + n;
    sXr[c][n] = Xre[g]; sXi[c][n] = Xim[g];
  }
  for (int idx = tid; idx < NSRC * NFR; idx += NT) {
    int c = idx / NFR, n = idx % NFR;
    size_t g = (((size_t)b * NSRC + c) * NFRQ + f) * (size_t)NFR + n;
    sYr[c][n] = out[2 * g]; sYi[c][n] = out[2 * g + 1];
    sw[c][n] = ws[OFF_WGT + (size_t)(b * NSRC + c) * NFR + n];
  }
  for (int idx = tid; idx < NBAK * NFR; idx += NT) {
    size_t o = OFF_Z + ((size_t)bf * NBAK * NFR + idx) * 2;
    sZr[idx / NFR][idx % NFR] = ws[o];
    sZi[idx / NFR][idx % NFR] = ws[o + 1];
  }
  if (tid < NSRC * NCH) {
    size_t o = OFF_W + ((size_t)bf * NSRC * NCH + tid) * 2;
    (&sWr[0][0])[tid] = ws[o]; (&sWi[0][0])[tid] = ws[o + 1];
  }
  if (tid < NSRC * NCH * NTAPS) {
    size_t o = OFF_H + ((size_t)bf * NSRC * NCH * NTAPS + tid) * 2;
    (&sHr[0][0][0])[tid] = ws[o]; (&sHi[0][0][0])[tid] = ws[o + 1];
  }
  if (tid < NSRC * NSRC) {
    size_t o = OFF_A + ((size_t)bf * NSRC * NSRC + tid) * 2;
    (&sAr[0][0])[tid] = ws[o]; (&sAi[0][0])[tid] = ws[o + 1];
  }
  if (tid < NBAK * NSRC) {
    size_t o = OFF_J + ((size_t)bf * NBAK * NSRC + tid) * 2;
    (&sJr[0][0])[tid] = ws[o]; (&sJi[0][0])[tid] = ws[o + 1];
  }
  __syncthreads();

  // ================= bak loop: bak_update + mat_up2 =================
  for (int bak = 0; bak < NBAK; ++bak) {
    if (tid < 18) acc[tid] = 0.f;
    __syncthreads();
    {
      float ppr[NCH] = {0}, ppi[NCH] = {0}, pdd[NCH] = {0};
      for (int n = tid; n < NFR; n += NT) {
        float zr = sZr[bak][n] + sXr[NSRC + bak][n];
        float zi = sZi[bak][n] + sXi[NSRC + bak][n];
        float mz = zr * zr + zi * zi;
        #pragma unroll
        for (int c = 0; c < NCH; ++c) {
          float yr = (c < NSRC) ? sYr[c][n] : sZr[c - NSRC][n];
          float yi = (c < NSRC) ? sYi[c][n] : sZi[c - NSRC][n];
          float wc = (c < NSRC) ? sw[c][n] : 1.f;
          ppr[c] += wc * (yr * zr + yi * zi);
          ppi[c] += wc * (yi * zr - yr * zi);
          pdd[c] += wc * mz;
        }
      }
      #pragma unroll
      for (int c = 0; c < NCH; ++c) {
        blockReduceAdd(&acc[3 * c + 0], ppr[c]);
        blockReduceAdd(&acc[3 * c + 1], ppi[c]);
        blockReduceAdd(&acc[3 * c + 2], pdd[c]);
      }
    }
    __syncthreads();
    if (tid < NCH) { // g[c] = conj(sum_s J[bak][s] A2[s][c]); p,d scaled
      int c = tid;
      float gr = 0.f, gi = 0.f;
      for (int s = 0; s < NSRC; ++s) {
        float a2r, a2i;
        if (c < NSRC) { a2r = sAr[s][c]; a2i = sAi[s][c]; }
        else {
          a2r = 0.f; a2i = 0.f;
          for (int k = 0; k < NSRC; ++k) {
            a2r += sAr[s][k] * sWr[k][c] - sAi[s][k] * sWi[k][c];
            a2i += sAr[s][k] * sWi[k][c] + sAi[s][k] * sWr[k][c];
          }
        }
        gr += sJr[bak][s] * a2r - sJi[bak][s] * a2i;
        gi += sJr[bak][s] * a2i + sJi[bak][s] * a2r;
      }
      sgr[c] = gr; sgi[c] = -gi;
      spr[c] = acc[3 * c + 0] * NORM; spi[c] = acc[3 * c + 1] * NORM; sdd[c] = acc[3 * c + 2] * NORM;
    }
    __syncthreads();
    if (tid == 0) { // bak_update_vector
      float gdg = 0.f, gpr = 0.f, gpi = 0.f;
      for (int c = 0; c < NCH; ++c) {
        float di = 1.f / (sdd[c] + EPS);
        gdg += (sgr[c] * sgr[c] + sgi[c] * sgi[c]) * di;
        gpr += (sgr[c] * spr[c] + sgi[c] * spi[c]) * di;
        gpi += (sgr[c] * spi[c] - sgi[c] * spr[c]) * di;
      }
      float br = 1.f - gpr, bi = -gpi;
      float b1 = br * br + bi * bi;
      float a = b1 * gdg + TINY;
      float beta = (-b1 + sqrtf(b1 * b1 + 4.f * a)) / (2.f * a);
      float er, ei;
      if (b1 > EPS * EPS) { er = beta * br; ei = beta * bi; }
      else { er = 1.f / sqrtf(EPS + gdg); ei = 0.f; }
      for (int c = 0; c < NCH; ++c) {
        float nr = spr[c] - (er * sgr[c] - ei * sgi[c]);
        float ni = spi[c] - (er * sgi[c] + ei * sgr[c]);
        float di = 1.f / (sdd[c] + EPS);
        svr[c] = nr * di; svi[c] = ni * di;
      }
    }
    __syncthreads();
    for (int n = tid; n < NFR; n += NT) { // YZ -= v * Zs
      float zr = sZr[bak][n] + sXr[NSRC + bak][n];
      float zi = sZi[bak][n] + sXi[NSRC + bak][n];
      #pragma unroll
      for (int c = 0; c < NCH; ++c) {
        float ur = svr[c] * zr - svi[c] * zi;
        float ui = svr[c] * zi + svi[c] * zr;
        if (c < NSRC) { sYr[c][n] -= ur; sYi[c][n] -= ui; }
        else { sZr[c - NSRC][n] -= ur; sZi[c - NSRC][n] -= ui; }
      }
    }
    // mat_up2: g2 = J[bak]; v2 = v[:4] + W[:,4:]·v[4:]
    if (tid < NSRC) { sg2r[tid] = sJr[bak][tid]; sg2i[tid] = sJi[bak][tid]; }
    else if (tid < 2 * NSRC) {
      int s = tid - NSRC;
      float r = svr[s], q = svi[s];
      for (int e = 0; e < NBAK; ++e) {
        r += sWr[s][NSRC + e] * svr[NSRC + e] - sWi[s][NSRC + e] * svi[NSRC + e];
        q += sWr[s][NSRC + e] * svi[NSRC + e] + sWi[s][NSRC + e] * svr[NSRC + e];
      }
      sv2r[s] = r; sv2i[s] = q;
    }
    __syncthreads();
    if (tid < NSRC) { // gA[c] = sum_s g2[s] A[s][c]
      int c = tid; float r = 0.f, q = 0.f;
      for (int s = 0; s < NSRC; ++s) {
        r += sg2r[s] * sAr[s][c] - sg2i[s] * sAi[s][c];
        q += sg2r[s] * sAi[s][c] + sg2i[s] * sAr[s][c];
      }
      sgAr[c] = r; sgAi[c] = q;
    } else if (tid < 2 * NSRC) { // Av[s] = sum_c A[s][c] v2[c]
      int s = tid - NSRC; float r = 0.f, q = 0.f;
      for (int c = 0; c < NSRC; ++c) {
        r += sAr[s][c] * sv2r[c] - sAi[s][c] * sv2i[c];
        q += sAr[s][c] * sv2i[c] + sAi[s][c] * sv2r[c];
      }
      sAvr[s] = r; sAvi[s] = q;
    }
    __syncthreads();
    if (tid == 0) {
      float dr = 1.f, di2 = 0.f;
      for (int c = 0; c < NSRC; ++c) {
        dr -= sv2r[c] * sgAr[c] - sv2i[c] * sgAi[c];
        di2 -= sv2r[c] * sgAi[c] + sv2i[c] * sgAr[c];
      }
      float m = dr * dr + di2 * di2 + EPS;
      smul[0] = dr / m; smul[1] = -di2 / m;
    }
    __syncthreads();
    if (tid < 16) {
      int s = tid >> 2, d = tid & 3;
      sWr[s][d] -= svr[s] * sg2r[d] - svi[s] * sg2i[d];
      sWi[s][d] -= svr[s] * sg2i[d] + svi[s] * sg2r[d];
    } else if (tid < 24) {
      int e = (tid - 16) >> 2, d = tid & 3;
      sJr[e][d] -= svr[NSRC + e] * sg2r[d] - svi[NSRC + e] * sg2i[d];
      sJi[e][d] -= svr[NSRC + e] * sg2i[d] + svi[NSRC + e] * sg2r[d];
    } else if (tid >= 32 && tid < 48) {
      int s = (tid - 32) >> 2, c = tid & 3;
      float tr = sAvr[s] * sgAr[c] - sAvi[s] * sgAi[c];
      float ti = sAvr[s] * sgAi[c] + sAvi[s] * sgAr[c];
      sAr[s][c] += tr * smul[0] - ti * smul[1];
      sAi[s][c] += tr * smul[1] + ti * smul[0];
    }
    __syncthreads();
  }

  // ================= src loop: iss_type1 + mat_up1 =================
  for (int src = 0; src < NSRC; ++src) {
    if (tid < 12) acc[tid] = 0.f;
    __syncthreads();
    {
      float vnr[NSRC] = {0}, vni[NSRC] = {0}, vdd[NSRC] = {0};
      for (int n = tid; n < NFR; n += NT) {
        float yr = sYr[src][n], yi = sYi[src][n];
        float my = yr * yr + yi * yi;
        #pragma unroll
        for (int c = 0; c < NSRC; ++c) {
          float wc = sw[c][n];
          vnr[c] += wc * (sYr[c][n] * yr + sYi[c][n] * yi);
          vni[c] += wc * (sYi[c][n] * yr - sYr[c][n] * yi);
          vdd[c] += wc * my;
        }
      }
      #pragma unroll
      for (int c = 0; c < NSRC; ++c) {
        blockReduceAdd(&acc[3 * c + 0], vnr[c]);
        blockReduceAdd(&acc[3 * c + 1], vni[c]);
        blockReduceAdd(&acc[3 * c + 2], vdd[c]);
      }
    }
    __syncthreads();
    if (tid < NSRC) {
      int c = tid;
      float vd = acc[3 * c + 2] * NORM;
      if (c == src) { svr[c] = 1.f - 1.f / sqrtf(fmaxf(vd, EPS)); svi[c] = 0.f; }
      else {
        float dn = fmaxf(vd, EPS);
        svr[c] = acc[3 * c + 0] * NORM / dn;
        svi[c] = acc[3 * c + 1] * NORM / dn;
      }
    }
    __syncthreads();
    if (tid < NCH) { sWsr[tid] = sWr[src][tid]; sWsi[tid] = sWi[src][tid]; }
    else if (tid >= 8 && tid < 8 + NCH * NTAPS) {
      int q = tid - 8, d = q >> 2, t = q & 3;
      sHsr[q] = sHr[src][d][t]; sHsi[q] = sHi[src][d][t];
    } else if (tid >= 40 && tid < 40 + NSRC) { // u = A·v / (1 - v[src] + eps)
      int s = tid - 40;
      float r = 0.f, q = 0.f;
      for (int c = 0; c < NSRC; ++c) {
        r += sAr[s][c] * svr[c] - sAi[s][c] * svi[c];
        q += sAr[s][c] * svi[c] + sAi[s][c] * svr[c];
      }
      float dr = 1.f - svr[src] + EPS, di2 = -svi[src];
      float m = dr * dr + di2 * di2;
      sur[s] = (r * dr + q * di2) / m;
      sui[s] = (q * dr - r * di2) / m;
    }
    __syncthreads();
    for (int n = tid; n < NFR; n += NT) {
      float yr = sYr[src][n], yi = sYi[src][n];
      #pragma unroll
      for (int c = 0; c < NSRC; ++c) {
        sYr[c][n] -= svr[c] * yr - svi[c] * yi;
        sYi[c][n] -= svr[c] * yi + svi[c] * yr;
      }
    }
    if (tid < NSRC * NCH) {
      int c = tid / NCH, d = tid % NCH;
      sWr[c][d] -= svr[c] * sWsr[d] - svi[c] * sWsi[d];
      sWi[c][d] -= svr[c] * sWsi[d] + svi[c] * sWsr[d];
    }
    if (tid >= 32 && tid < 32 + NSRC * NCH * NTAPS) {
      int q = tid - 32, c = q / (NCH * NTAPS), r2 = q % (NCH * NTAPS);
      int d = r2 >> 2, t = r2 & 3;
      sHr[c][d][t] -= svr[c] * sHsr[r2] - svi[c] * sHsi[r2];
      sHi[c][d][t] -= svr[c] * sHsi[r2] + svi[c] * sHsr[r2];
    }
    if (tid >= 128 && tid < 128 + NSRC) {
      int s = tid - 128;
      sAr[s][src] += sur[s]; sAi[s][src] += sui[s];
    }
    __syncthreads();
  }

  // ================= type2 loop =================
  for (int bak = 0; bak < NBAK; ++bak) {
    if (tid < 12) acc[tid] = 0.f;
    __syncthreads();
    {
      float vnr[NSRC] = {0}, vni[NSRC] = {0}, vdd[NSRC] = {0};
      for (int n = tid; n < NFR; n += NT) {
        float zr = sZr[bak][n], zi = sZi[bak][n];
        float mz = zr * zr + zi * zi;
        #pragma unroll
        for (int c = 0; c < NSRC; ++c) {
          float wc = sw[c][n];
          vnr[c] += wc * (sYr[c][n] * zr + sYi[c][n] * zi);
          vni[c] += wc * (sYi[c][n] * zr - sYr[c][n] * zi);
          vdd[c] += wc * mz;
        }
      }
      #pragma unroll
      for (int c = 0; c < NSRC; ++c) {
        blockReduceAdd(&acc[3 * c + 0], vnr[c]);
        blockReduceAdd(&acc[3 * c + 1], vni[c]);
        blockReduceAdd(&acc[3 * c + 2], vdd[c]);
      }
    }
    __syncthreads();
    if (tid < NSRC) {
      float dn = fmaxf(acc[3 * tid + 2], EPS);
      svr[tid] = acc[3 * tid + 0] / dn;
      svi[tid] = acc[3 * tid + 1] / dn;
    }
    __syncthreads();
    for (int n = tid; n < NFR; n += NT) {
      float zr = sZr[bak][n], zi = sZi[bak][n];
      #pragma unroll
      for (int c = 0; c < NSRC; ++c) {
        sYr[c][n] -= svr[c] * zr - svi[c] * zi;
        sYi[c][n] -= svr[c] * zi + svi[c] * zr;
      }
    }
    if (tid < 16) {
      int c = tid >> 2, d = tid & 3;
      sWr[c][d] -= svr[c] * sJr[bak][d] - svi[c] * sJi[bak][d];
      sWi[c][d] -= svr[c] * sJi[bak][d] + svi[c] * sJr[bak][d];
    } else if (tid < 20) {
      int c = tid - 16;
      sWr[c][NSRC + bak] += svr[c];
      sWi[c][NSRC + bak] += svi[c];
    }
    __syncthreads();
  }

  // ================= Binv = W[:, :4] + W[:,4:]·J ; A = Binv^-1 =================
  if (tid < 16) {
    int s = tid >> 2, d = tid & 3;
    float r = sWr[s][d], q = sWi[s][d];
    for (int e = 0; e < NBAK; ++e) {
      r += sWr[s][NSRC + e] * sJr[e][d] - sWi[s][NSRC + e] * sJi[e][d];
      q += sWr[s][NSRC + e] * sJi[e][d] + sWi[s][NSRC + e] * sJr[e][d];
    }
    sBr[s][d] = r; sBi[s][d] = q;
  }
  __syncthreads();
  if (tid == 0) {
    cpx M[4][8];
    for (int i = 0; i < 4; ++i)
      for (int j = 0; j < 4; ++j) {
        M[i][j] = {sBr[i][j], sBi[i][j]};
        M[i][4 + j] = {(i == j) ? 1.f : 0.f, 0.f};
      }
    for (int k = 0; k < 4; ++k) {
      int p = k; float best = M[k][k].r * M[k][k].r + M[k][k].i * M[k][k].i;
      for (int r = k + 1; r < 4; ++r) {
        float mg = M[r][k].r * M[r][k].r + M[r][k].i * M[r][k].i;
        if (mg > best) { best = mg; p = r; }
      }
      if (p != k) for (int j = 0; j < 8; ++j) { cpx t = M[k][j]; M[k][j] = M[p][j]; M[p][j] = t; }
      cpx inv = crec(M[k][k]);
      for (int j = 0; j < 8; ++j) M[k][j] = cmul(M[k][j], inv);
      for (int r = 0; r < 4; ++r) if (r != k) {
        cpx fc = M[r][k];
        for (int j = 0; j < 8; ++j) M[r][j] = csub(M[r][j], cmul(fc, M[k][j]));
      }
    }
    for (int i = 0; i < 4; ++i)
      for (int j = 0; j < 4; ++j) { sAr[i][j] = M[i][4 + j].r; sAi[i][j] = M[i][4 + j].i; }
  }
  __syncthreads();

  // ================= type3 loop: dereverberation taps (X_bar from shifted X) ==
  for (int src = 0; src < NCH; ++src) {
    for (int tap = 0; tap < NTAPS; ++tap) {
      const int sh = 1 + tap; // N_DELAY + tap
      if (tid < 12) acc[tid] = 0.f;
      __syncthreads();
      {
        float vnr[NSRC] = {0}, vni[NSRC] = {0}, vdd[NSRC] = {0};
        for (int n = tid; n < NFR; n += NT) {
          float xr = (n >= sh) ? sXr[src][n - sh] : 0.f;
          float xi = (n >= sh) ? sXi[src][n - sh] : 0.f;
          float mx = xr * xr + xi * xi;
          #pragma unroll
          for (int c = 0; c < NSRC; ++c) {
            float wc = sw[c][n];
            vnr[c] += wc * (sYr[c][n] * xr + sYi[c][n] * xi);
            vni[c] += wc * (sYi[c][n] * xr - sYr[c][n] * xi);
            vdd[c] += wc * mx;
          }
        }
        #pragma unroll
        for (int c = 0; c < NSRC; ++c) {
          blockReduceAdd(&acc[3 * c + 0], vnr[c]);
          blockReduceAdd(&acc[3 * c + 1], vni[c]);
          blockReduceAdd(&acc[3 * c + 2], vdd[c]);
        }
      }
      __syncthreads();
      if (tid < NSRC) {
        float dn = fmaxf(acc[3 * tid + 2], EPS);
        svr[tid] = acc[3 * tid + 0] / dn;
        svi[tid] = acc[3 * tid + 1] / dn;
      }
      __syncthreads();
      for (int n = tid; n < NFR; n += NT) {
        if (n >= sh) {
          float xr = sXr[src][n - sh], xi = sXi[src][n - sh];
          #pragma unroll
          for (int c = 0; c < NSRC; ++c) {
            sYr[c][n] -= svr[c] * xr - svi[c] * xi;
            sYi[c][n] -= svr[c] * xi + svi[c] * xr;
          }
        }
      }
      if (tid < NSRC) {
        sHr[tid][src][tap] += svr[tid];
        sHi[tid][src][tap] += svi[tid];
      }
      __syncthreads();
    }
  }

  // ---- write back state ----
  for (int idx = tid; idx < NSRC * NFR; idx += NT) {
    int c = idx / NFR, n = idx % NFR;
    size_t g = (((size_t)b * NSRC + c) * NFRQ + f) * (size_t)NFR + n;
    out[2 * g] = sYr[c][n]; out[2 * g + 1] = sYi[c][n];
  }
  for (int idx = tid; idx < NBAK * NFR; idx += NT) {
    size_t o = OFF_Z + ((size_t)bf * NBAK * NFR + idx) * 2;
    ws[o] = sZr[idx / NFR][idx % NFR];
    ws[o + 1] = sZi[idx / NFR][idx % NFR];
  }
  if (tid < NSRC * NCH) {
    size_t o = OFF_W + ((size_t)bf * NSRC * NCH + tid) * 2;
    ws[o] = (&sWr[0][0])[tid]; ws[o + 1] = (&sWi[0][0])[tid];
  }
  if (tid < NSRC * NCH * NTAPS) {
    size_t o = OFF_H + ((size_t)bf * NSRC * NCH * NTAPS + tid) * 2;
    ws[o] = (&sHr[0][0][0])[tid]; ws[o + 1] = (&sHi[0][0][0])[tid];
  }
  if (tid < NSRC * NSRC) {
    size_t o = OFF_A + ((size_t)bf * NSRC * NSRC + tid) * 2;
    ws[o] = (&sAr[0][0])[tid]; ws[o + 1] = (&sAi[0][0])[tid];
  }
  if (tid < NBAK * NSRC) {
    size_t o = OFF_J + ((size_t)bf * NBAK * NSRC + tid) * 2;
    ws[o] = (&sJr[0][0])[tid]; ws[o + 1] = (&sJi[0][0])[tid];
  }
}

extern "C" void kernel_launch(void* const* d_in, const int* in_sizes, int n_in,
                              void* d_out, int out_size, void* d_ws, size_t ws_size,
                              hipStream_t stream) {
  (void)in_sizes; (void)n_in; (void)out_size; (void)ws_size;
  const float* Xre = (const float*)d_in[0];
  const float* Xim = (const float*)d_in[1];
  float* out = (float*)d_out;
  float* ws = (float*)d_ws;

  init_kernel<<<dim3(NBF), dim3(NT), 0, stream>>>(Xre, Xim, out, ws);
  for (int it = 0; it < 2; ++it) {
    weight_kernel<<<dim3(NB * NSRC), dim3(NT), 0, stream>>>(out, ws);
    iss_kernel<<<dim3(NBF), dim3(NT), 0, stream>>>(Xre, Xim, out, ws);
  }
}